// SupermajorityFilterLoss_44006234915308
// MI455X (gfx1250) — compile-verified
//
#include <hip/hip_runtime.h>
#include <stdint.h>

#define L_LEN  16384
#define CHUNK  64
#define NCHUNK 256
#define BIGF   1e30f

typedef float v2f __attribute__((ext_vector_type(2)));
typedef float v8f __attribute__((ext_vector_type(8)));

// LDS byte offset (AS3 pointers are 32-bit offsets from the workgroup LDS base,
// which is exactly what global_load_async_to_lds_* wants in its VDST VGPR).
// Guarded: the hipcc host pass also sema-checks this body, where pointers are
// 64-bit, so keep the addrspacecast device-only and narrow via uintptr_t.
__device__ __forceinline__ unsigned lds_off(const void* p) {
#if defined(__HIP_DEVICE_COMPILE__)
    return (unsigned)(uintptr_t)(__attribute__((address_space(3))) const void*)p;
#else
    return 0u;
#endif
}

// Swizzled transposed slot for element e (0..63) of chunk c (0..255):
//   transposed  -> compute-phase reads by lane c are consecutive (bank-free)
//   +e rotation -> copy-phase async LDS writes by a wave are also bank-free
__device__ __forceinline__ unsigned slot(int e, int c) {
    return (unsigned)((e << 8) | ((c + e) & 255));
}

__global__ __launch_bounds__(256) void viterbi_rows(const float* __restrict__ xg,
                                                    const int*   __restrict__ tg,
                                                    float*       __restrict__ row_min) {
    extern __shared__ char smem[];
    float*         sX = (float*)smem;                          // 64 KB
    unsigned char* sT = (unsigned char*)(smem + 65536);        // 16 KB
    float*         sM = (float*)(smem + 65536 + 16384);        // 16 KB (256 x 4x4)

    const int tid = threadIdx.x;
    const int row = blockIdx.x;
    const float* gx = xg + (size_t)row * L_LEN;
    const int*   gt = tg + (size_t)row * L_LEN;

    const unsigned baseX = lds_off(sX);
    const unsigned baseT = lds_off(sT);

    // ---- Stage the whole row into LDS with CDNA5 async copies (transposed) ----
    #pragma unroll 4
    for (int it = 0; it < 64; ++it) {
        int g = (it << 8) + tid;          // element index within the row
        int c = g >> 6;
        int e = g & 63;
        unsigned s  = slot(e, c);
        unsigned dX = baseX + 4u * s;
        unsigned long long aX = (unsigned long long)(uintptr_t)(gx + g);
        asm volatile("global_load_async_to_lds_b32 %0, %1, off"
                     :: "v"(dX), "v"(aX) : "memory");
        unsigned dT = baseT + s;          // LSB byte of the little-endian int is the 0/1 value
        unsigned long long aT = (unsigned long long)(uintptr_t)(gt + g);
        asm volatile("global_load_async_to_lds_b8 %0, %1, off"
                     :: "v"(dT), "v"(aT) : "memory");
    }
    asm volatile("s_wait_asynccnt 0" ::: "memory");
    __syncthreads();

    // ---- Per-lane: min-plus 4x4 transfer matrix of this lane's 64-element chunk ----
    const int c  = tid;
    const int i0 = c << 6;

    float A[4][4];
    #pragma unroll
    for (int s = 0; s < 4; ++s)
        #pragma unroll
        for (int j = 0; j < 4; ++j)
            A[s][j] = (s == j) ? 0.0f : BIGF;

    auto loadT = [&](int i) -> int {
        if ((unsigned)i >= (unsigned)L_LEN) return 0;   // outside row behaves as target==0
        return (int)sT[slot(i & 63, i >> 6)];
    };

    int tm2 = loadT(i0 - 2);
    int tm1 = loadT(i0 - 1);
    int tc  = loadT(i0);
    int tn  = loadT(i0 + 1);

    #pragma unroll 2
    for (int e = 0; e < CHUNK; ++e) {
        float x  = sX[slot(e, c)];
        // softplus(x) stably; softplus(-x) = softplus(x) - x
        float zl = fmaxf(x, 0.0f) + logf(1.0f + expf(-fabsf(x)));   // cost of bit 0
        float ol = zl - x;                                          // cost of bit 1

        bool is1  = tc != 0;
        bool tiny = is1 && (tm1 == 0) && (tn == 0);                 // rlen==1 -> force 1
        bool sh   = is1 && (tm1 != 0) && (tm2 == 0) && (tn == 0);   // rlen==2, pos==1
        bool fu   = is1 && (tm1 != 0) && (tm2 != 0);                // full 3-window ends here

        // allowed(pred state s, emitted bit b) -> penalized emission costs
        float P0[4], P1[4];
        #pragma unroll
        for (int s = 0; s < 4; ++s) {
            bool a0 = !is1 || (!(tiny || sh) && (!fu || s == 3));           // b=0
            bool a1 = is1 && (tiny || (sh ? ((s & 1) != 0) : (!fu || s != 0))); // b=1
            P0[s] = a0 ? zl : BIGF;
            P1[s] = a1 ? ol : BIGF;
        }

        // ns = ((s&1)<<1) | b  ;  preds of ns: {ns>>1, (ns>>1)|2}, emitted bit ns&1
        float N[4][4];
        #pragma unroll
        for (int ns = 0; ns < 4; ++ns) {
            int p0 = ns >> 1, p1 = p0 | 2;
            float e0 = (ns & 1) ? P1[p0] : P0[p0];
            float e1 = (ns & 1) ? P1[p1] : P0[p1];
            #pragma unroll
            for (int j = 0; j < 4; ++j)
                N[ns][j] = fminf(A[p0][j] + e0, A[p1][j] + e1);
        }
        #pragma unroll
        for (int s = 0; s < 4; ++s)
            #pragma unroll
            for (int j = 0; j < 4; ++j)
                A[s][j] = N[s][j];

        tm2 = tm1; tm1 = tc; tc = tn;
        tn = loadT(i0 + e + 2);
    }

    // ---- Log-tree min-plus composition of the 256 chunk matrices ----
    #pragma unroll
    for (int s = 0; s < 4; ++s)
        #pragma unroll
        for (int j = 0; j < 4; ++j)
            sM[tid * 16 + s * 4 + j] = A[s][j];
    __syncthreads();

    for (int stride = 1; stride < NCHUNK; stride <<= 1) {
        if ((tid & (2 * stride - 1)) == 0) {
            const float* Lm = &sM[tid * 16];                // earlier chunks
            const float* Rm = &sM[(tid + stride) * 16];     // later chunks
            float T[16];
            #pragma unroll
            for (int i = 0; i < 4; ++i)
                #pragma unroll
                for (int j = 0; j < 4; ++j) {
                    float v = Rm[i * 4 + 0] + Lm[0 * 4 + j];
                    v = fminf(v, Rm[i * 4 + 1] + Lm[1 * 4 + j]);
                    v = fminf(v, Rm[i * 4 + 2] + Lm[2 * 4 + j]);
                    v = fminf(v, Rm[i * 4 + 3] + Lm[3 * 4 + j]);
                    T[i * 4 + j] = v;
                }
            #pragma unroll
            for (int k = 0; k < 16; ++k) sM[tid * 16 + k] = T[k];
        }
        __syncthreads();
    }

    if (tid == 0) {
        // dp0 = BIG except state 0 -> row loss = min_s M[s][0]
        float v = fminf(fminf(sM[0], sM[4]), fminf(sM[8], sM[12]));
        row_min[row] = v;
    }
}

// Final mean over the per-row losses, done on the matrix pipe:
// V_WMMA_F32_16X16X4_F32 with B == ones gives D[i][j] = sum_k A[i][k] (+C),
// so 4 accumulating WMMAs sum 256 f32 values into column 0 of D (exact f32,
// fixed hardware order -> deterministic). Single wave, EXEC all-ones.
__global__ __launch_bounds__(32) void reduce_mean_wmma(const float* __restrict__ rows,
                                                       float* __restrict__ out,
                                                       int nrows, float scale) {
    const int lane = threadIdx.x;
    const int chunks = nrows >> 6;            // 64 values per WMMA (256 -> 4)

    v2f ones; ones.x = 1.0f; ones.y = 1.0f;   // B: 4x16 all-ones
    v8f acc = {};                              // C/D accumulator

    // A layout (16x4 f32): lanes 0-15 hold (M=lane, K=0/1) in v0/v1,
    // lanes 16-31 hold (M=lane-16, K=2/3). Any bijection of the 64 inputs
    // onto (M,K) works since B is all-ones; use idx = M*4 + K.
    const int m  = lane & 15;
    const int k0 = (lane < 16) ? 0 : 2;

    for (int cnk = 0; cnk < chunks; ++cnk) {
        const int base = cnk << 6;
        v2f a;
        a.x = rows[base + m * 4 + k0];
        a.y = rows[base + m * 4 + k0 + 1];
        acc = __builtin_amdgcn_wmma_f32_16x16x4_f32(
                  /*neg_a=*/false, a, /*neg_b=*/false, ones,
                  /*c_mod=*/(short)0, acc, /*reuse_a=*/false, /*reuse_b=*/false);
    }

    // Lane L holds D rows M = {0..7 | 8..15 for L>=16}, column N = L%16.
    // Sum over i of D[i][0] = (reg-sum in lane 0) + (reg-sum in lane 16).
    float s = acc[0] + acc[1] + acc[2] + acc[3] + acc[4] + acc[5] + acc[6] + acc[7];

    // Scalar tail if nrows is not a multiple of 64 (not hit for B=256).
    float tail = 0.0f;
    if (lane == 0)
        for (int i = chunks << 6; i < nrows; ++i) tail += rows[i];

    __shared__ float sh[32];
    sh[lane] = s;
    __syncthreads();
    if (lane == 0) out[0] = (sh[0] + sh[16] + tail) * scale;
}

extern "C" void kernel_launch(void* const* d_in, const int* in_sizes, int n_in,
                              void* d_out, int out_size, void* d_ws, size_t ws_size,
                              hipStream_t stream) {
    const float* x = (const float*)d_in[0];
    const int*   t = (const int*)d_in[1];
    const int L = L_LEN;
    const int B = in_sizes[0] / L;          // 256 for the reference shapes
    float* rows = (float*)d_ws;             // B floats of scratch

    size_t smem = (size_t)65536 + 16384 + 16384;   // 96 KB dynamic LDS (<= 320 KB/WGP)
    viterbi_rows<<<B, 256, smem, stream>>>(x, t, rows);
    reduce_mean_wmma<<<1, 32, 0, stream>>>(rows, (float*)d_out, B,
                                           1.0f / ((float)B * (float)L));
}